// SimpleAggregation_30786325578421
// MI455X (gfx1250) — compile-verified
//
#include <hip/hip_runtime.h>
#include <stdint.h>

// ---------------------------------------------------------------------------
// Segment max + mean over sorted segment ids, MI455X (gfx1250).
// Memory-bound (~264 MB traffic -> ~11us floor @ 23.3 TB/s). Strategy:
//   * 1 block (1 wave32) per segment.
//   * Binary-search the sorted id array for the segment's contiguous row range.
//   * Tensor Data Mover DMAs each <=32-row x 128-f32 tile (16 KB) into LDS,
//     double-buffered, tracked with TENSORcnt (s_wait_tensorcnt).
//   * Wave reduces from LDS: lane L owns float4 columns [4L..4L+3].
// ---------------------------------------------------------------------------

typedef __attribute__((ext_vector_type(4))) unsigned int v4u;
typedef __attribute__((ext_vector_type(8))) unsigned int v8u;

#define DCOLS      128
#define CHUNK_ROWS 32

__device__ __forceinline__ int lower_bound_dev(const int* __restrict__ ids, int m, int key) {
    int lo = 0, hi = m;
    while (lo < hi) {
        int mid = (lo + hi) >> 1;
        if (ids[mid] < key) lo = mid + 1; else hi = mid;
    }
    return lo;
}

template <int N>
__device__ __forceinline__ void wait_tensorcnt() {
#if __has_builtin(__builtin_amdgcn_s_wait_tensorcnt)
    __builtin_amdgcn_s_wait_tensorcnt(N);
#else
    asm volatile("s_wait_tensorcnt %0" :: "i"(N) : "memory");
#endif
}

// Issue one TDM load: `rows` x 128-f32 tile at gbase -> LDS buffer at lds_addr.
// Descriptor layout per CDNA5 ISA (2-D, groups 0+1).
__device__ __forceinline__ void tdm_load_tile(const float* gbase,
                                              unsigned lds_addr, int rows) {
    unsigned long long gaddr = (unsigned long long)(uintptr_t)gbase;

    // ---- D# group 0 (128b): count=1, lds_addr, global_addr, type=2 ----
    v4u g0;
    g0[0] = 1u;                                            // count=1, user mode
    g0[1] = lds_addr;                                      // LDS byte address
    g0[2] = (unsigned)(gaddr & 0xFFFFFFFFu);               // global_addr[31:0]
    g0[3] = (unsigned)((gaddr >> 32) & 0x01FFFFFFu)        // global_addr[56:32]
          | (2u << 30);                                    // type=2 ("image")

    // ---- D# group 1 (256b): 2-D tile descriptor ----
    v8u g1;
    g1[0] = 2u << 16;                                      // data_size=2 (4 bytes)
    g1[1] = (unsigned)DCOLS << 16;                         // tensor_dim0[15:0] -> dw1[31:16]
    g1[2] = ((unsigned)rows & 0xFFFFu) << 16;              // tensor_dim0 hi=0 | tensor_dim1[15:0]=rows
    g1[3] = (unsigned)DCOLS << 16;                         // tensor_dim1 hi=0 | tile_dim0=128
    g1[4] = (unsigned)rows & 0xFFFFu;                      // tile_dim1=rows | tile_dim2=0
    g1[5] = (unsigned)DCOLS;                               // tensor_dim0_stride[31:0] = 128 elems
    g1[6] = 0u;                                            // stride0 hi | stride1 lo (unused, 2-D)
    g1[7] = 0u;

    // Two-operand form: VADDR2/VADDR3 = NULL (<=2-D tensor).
    asm volatile("tensor_load_to_lds %0, %1" :: "s"(g0), "s"(g1) : "memory");
}

__global__ __launch_bounds__(32)
void SegAgg_tdm_kernel(const float* __restrict__ lane_enc,
                       const int*   __restrict__ ids,
                       float*       __restrict__ out,
                       int M) {
    __shared__ float4 sbuf[2][CHUNK_ROWS * (DCOLS / 4)];   // 2 x 16 KB double buffer

    const int seg  = blockIdx.x;
    const int lane = threadIdx.x;                          // 0..31, owns 4 columns

    // Contiguous row range of this segment (ids sorted ascending).
    int start = lower_bound_dev(ids, M, seg);
    int end   = lower_bound_dev(ids, M, seg + 1);
    start = __builtin_amdgcn_readfirstlane(start);         // force SGPR-uniform for D#
    end   = __builtin_amdgcn_readfirstlane(end);
    const int count   = end - start;
    const int nchunks = (count + CHUNK_ROWS - 1) / CHUNK_ROWS;

    // Wave-relative LDS byte addresses (generic addr truncates to LDS offset).
    const unsigned lds0 = (unsigned)(unsigned long long)(uintptr_t)(void*)&sbuf[0][0];
    const unsigned lds1 = (unsigned)(unsigned long long)(uintptr_t)(void*)&sbuf[1][0];

    float4 mx, sm;
    mx.x = mx.y = mx.z = mx.w = -__builtin_inff();
    sm.x = sm.y = sm.z = sm.w = 0.0f;

    if (nchunks > 0) {
        tdm_load_tile(lane_enc + (size_t)start * DCOLS, lds0, min(count, CHUNK_ROWS));
    }

    for (int c = 0; c < nchunks; ++c) {
        const int rows_c = min(count - c * CHUNK_ROWS, CHUNK_ROWS);
        if (c + 1 < nchunks) {
            const int rows_n = min(count - (c + 1) * CHUNK_ROWS, CHUNK_ROWS);
            tdm_load_tile(lane_enc + (size_t)(start + (c + 1) * CHUNK_ROWS) * DCOLS,
                          ((c + 1) & 1) ? lds1 : lds0, rows_n);
            wait_tensorcnt<1>();                           // chunk c complete (in-order)
        } else {
            wait_tensorcnt<0>();
        }
        asm volatile("" ::: "memory");                     // TDM wrote LDS: force reload

        const float4* __restrict__ buf = &sbuf[c & 1][0];
#pragma unroll 4
        for (int r = 0; r < rows_c; ++r) {
            float4 v = buf[r * (DCOLS / 4) + lane];        // ds_load_b128
            mx.x = fmaxf(mx.x, v.x);  sm.x += v.x;
            mx.y = fmaxf(mx.y, v.y);  sm.y += v.y;
            mx.z = fmaxf(mx.z, v.z);  sm.z += v.z;
            mx.w = fmaxf(mx.w, v.w);  sm.w += v.w;
        }
        asm volatile("" ::: "memory");                     // reads done before buffer reuse
    }

    // mean = sum / count  (count==0 -> 0/0 = NaN, matching JAX segment semantics)
    const float fc = (float)count;
    float4 mean;
    mean.x = sm.x / fc;  mean.y = sm.y / fc;
    mean.z = sm.z / fc;  mean.w = sm.w / fc;

    float* o = out + (size_t)seg * (2 * DCOLS);
    ((float4*)o)[lane]           = mx;                     // columns [0,128): seg_max
    ((float4*)(o + DCOLS))[lane] = mean;                   // columns [128,256): seg_mean
}

extern "C" void kernel_launch(void* const* d_in, const int* in_sizes, int n_in,
                              void* d_out, int out_size, void* d_ws, size_t ws_size,
                              hipStream_t stream) {
    // Inputs (reference order): obs_encoding (unused), lane_encoding, same_obs_mask.
    const float* lane_enc = (const float*)d_in[1];
    const int*   ids      = (const int*)d_in[2];
    float*       out      = (float*)d_out;

    const int M = in_sizes[2];              // 524288 rows
    const int N = in_sizes[0] / DCOLS;      // 8192 segments

    SegAgg_tdm_kernel<<<N, 32, 0, stream>>>(lane_enc, ids, out, M);
}